// GruSequenceToVector_3917010174692
// MI455X (gfx1250) — compile-verified
//
#include <hip/hip_runtime.h>
#include <hip/hip_bf16.h>

typedef float v2f __attribute__((ext_vector_type(2)));
typedef float v8f __attribute__((ext_vector_type(8)));

#define D     128
#define NTOK  1024           // B*S
#define G3    384            // 3*D

__device__ __forceinline__ v8f wmma4(v2f a, v2f b, v8f c) {
    // V_WMMA_F32_16X16X4_F32 : D(16x16,f32) = A(16x4,f32) * B(4x16,f32) + C
    return __builtin_amdgcn_wmma_f32_16x16x4_f32(
        /*neg_a=*/false, a, /*neg_b=*/false, b,
        /*c_mod=*/(short)0, c, /*reuse_a=*/false, /*reuse_b=*/false);
}

__device__ __forceinline__ float sigmoidf_dev(float x) {
    return 1.0f / (1.0f + __expf(-x));
}
__device__ __forceinline__ float tanhf_dev(float x) {
    x = fminf(15.0f, fmaxf(-15.0f, x));
    const float e = __expf(-2.0f * x);
    return (1.0f - e) / (1.0f + e);
}

// ---------------------------------------------------------------------------
// Phase 1 (parallel, device-wide): Xproj[t][j] = sum_k xs[t][k]*Wx[j][k] + bx[j]
//   (1024 x 384) = (1024 x 128) @ (128 x 384);  64 x 24 = 1536 16x16 tiles.
//   256 threads = 8 waves per block, one tile per wave, 192 blocks.
// ---------------------------------------------------------------------------
__global__ __launch_bounds__(256) void gru_xproj_kernel(
    const float* __restrict__ xs, const float* __restrict__ Wx,
    const float* __restrict__ bx, float* __restrict__ Xp)
{
    const int lane = threadIdx.x & 31;
    const int wave = threadIdx.x >> 5;
    const int tile = blockIdx.x * 8 + wave;     // 0..1535
    const int tm = tile / 24;                   // token tile  (0..63)
    const int tn = tile % 24;                   // column tile (0..23)
    const int l15 = lane & 15;
    const int hi  = lane >> 4;

    const float* __restrict__ arow = &xs[(tm * 16 + l15) * D];     // A: 16x4 slice of xs
    const float* __restrict__ brow = &Wx[(tn * 16 + l15) * D];     // B: column = Wx row

    v8f acc = {};
    for (int k0 = 0; k0 < D; k0 += 4) {
        const int kb = k0 + 2 * hi;
        v2f a = { arow[kb], arow[kb + 1] };
        v2f b = { brow[kb], brow[kb + 1] };
        acc = wmma4(a, b, acc);
    }

    const int col  = tn * 16 + l15;
    const float bb = bx[col];
#pragma unroll
    for (int r = 0; r < 8; ++r) {
        const int row = tm * 16 + r + 8 * hi;   // C/D layout: VGPR r, hi-half -> +8
        Xp[row * G3 + col] = acc[r] + bb;
    }
}

// ---------------------------------------------------------------------------
// Phase 2 (serial recurrence): one persistent 1024-thread block (32 waves).
// h1,h2 (128x128 f32 each) live in LDS. Per step:
//   layer1: Yh = h1 @ Wh^T (3 gate GEMMs, shared A), gates w/ hoisted Xp[t]
//   layer2: Yh = h2 @ Wh^T and Yx = h1_new @ Wx^T (6 accumulators, 2 shared A)
// 64 tile-groups (8 row-tiles x 8 col-tiles), 2 per wave.
// ---------------------------------------------------------------------------
__global__ __launch_bounds__(1024) void gru_serial_kernel(
    const int*   __restrict__ mask,
    const float* __restrict__ Wh, const float* __restrict__ bh,
    const float* __restrict__ Wx, const float* __restrict__ bx,
    const float* __restrict__ Xp,
    float* __restrict__ out)
{
    __shared__ float h1[D * D];   // 64 KB
    __shared__ float h2[D * D];   // 64 KB  (320 KB/WGP on CDNA5 -> fits easily)

    const int tid  = threadIdx.x;
    const int lane = tid & 31;
    const int wave = tid >> 5;
    const int l15  = lane & 15;
    const int hi   = lane >> 4;

    for (int i = tid; i < D * D; i += 1024) { h1[i] = 0.0f; h2[i] = 0.0f; }
    __syncthreads();

    for (int t = 0; t < NTOK; ++t) {
        const int mv = __builtin_amdgcn_readfirstlane(mask[t]);
        if (mv == 0) continue;   // masked token: both layers' states unchanged

        // ================= layer 1 =================
        float nv1[2][8];
#pragma unroll
        for (int gi = 0; gi < 2; ++gi) {
            const int g  = wave + gi * 32;   // 0..63
            const int mt = g >> 3;           // row tile 0..7
            const int jt = g & 7;            // col tile 0..7
            const float* __restrict__ arow = &h1[(mt * 16 + l15) * D];
            const float* __restrict__ wA = &Wh[(0 * D + jt * 16 + l15) * D];
            const float* __restrict__ wB = &Wh[(1 * D + jt * 16 + l15) * D];
            const float* __restrict__ wC = &Wh[(2 * D + jt * 16 + l15) * D];

            v8f accA = {}, accB = {}, accC = {};
            for (int k0 = 0; k0 < D; k0 += 4) {
                const int kb = k0 + 2 * hi;
                v2f a  = { arow[kb], arow[kb + 1] };      // A reused by 3 gates
                v2f b0 = { wA[kb], wA[kb + 1] };
                v2f b1 = { wB[kb], wB[kb + 1] };
                v2f b2 = { wC[kb], wC[kb + 1] };
                accA = wmma4(a, b0, accA);
                accB = wmma4(a, b1, accB);
                accC = wmma4(a, b2, accC);
            }

            const int col = jt * 16 + l15;
            const float sa = Xp[t * G3 + col]           + bh[col];
            const float sb = Xp[t * G3 + D + col]       + bh[D + col];
            const float xc = Xp[t * G3 + 2 * D + col];
            const float bc = bh[2 * D + col];
#pragma unroll
            for (int r = 0; r < 8; ++r) {
                const int row = mt * 16 + r + 8 * hi;
                const float u  = sigmoidf_dev(accA[r] + sa);
                const float rg = sigmoidf_dev(accB[r] + sb);
                const float cd = tanhf_dev(xc + rg * (accC[r] + bc));
                const float ho = h1[row * D + col];
                nv1[gi][r] = u * ho + (1.0f - u) * cd;
            }
        }
        __syncthreads();                      // all reads of old h1 done
#pragma unroll
        for (int gi = 0; gi < 2; ++gi) {
            const int g = wave + gi * 32;
            const int mt = g >> 3, jt = g & 7;
            const int col = jt * 16 + l15;
#pragma unroll
            for (int r = 0; r < 8; ++r)
                h1[(mt * 16 + r + 8 * hi) * D + col] = nv1[gi][r];
        }
        __syncthreads();                      // h1_new visible -> layer 2 input

        // ================= layer 2 =================
        float nv2[2][8];
#pragma unroll
        for (int gi = 0; gi < 2; ++gi) {
            const int g  = wave + gi * 32;
            const int mt = g >> 3;
            const int jt = g & 7;
            const float* __restrict__ aH = &h2[(mt * 16 + l15) * D];
            const float* __restrict__ aX = &h1[(mt * 16 + l15) * D];
            const float* __restrict__ whA = &Wh[(0 * D + jt * 16 + l15) * D];
            const float* __restrict__ whB = &Wh[(1 * D + jt * 16 + l15) * D];
            const float* __restrict__ whC = &Wh[(2 * D + jt * 16 + l15) * D];
            const float* __restrict__ wxA = &Wx[(0 * D + jt * 16 + l15) * D];
            const float* __restrict__ wxB = &Wx[(1 * D + jt * 16 + l15) * D];
            const float* __restrict__ wxC = &Wx[(2 * D + jt * 16 + l15) * D];

            v8f hA = {}, hB = {}, hC = {}, xA = {}, xB = {}, xC = {};
            for (int k0 = 0; k0 < D; k0 += 4) {
                const int kb = k0 + 2 * hi;
                v2f a2 = { aH[kb], aH[kb + 1] };
                v2f a1 = { aX[kb], aX[kb + 1] };
                v2f bh0 = { whA[kb], whA[kb + 1] };
                v2f bh1 = { whB[kb], whB[kb + 1] };
                v2f bh2 = { whC[kb], whC[kb + 1] };
                v2f bx0 = { wxA[kb], wxA[kb + 1] };
                v2f bx1 = { wxB[kb], wxB[kb + 1] };
                v2f bx2 = { wxC[kb], wxC[kb + 1] };
                hA = wmma4(a2, bh0, hA);
                hB = wmma4(a2, bh1, hB);
                hC = wmma4(a2, bh2, hC);
                xA = wmma4(a1, bx0, xA);
                xB = wmma4(a1, bx1, xB);
                xC = wmma4(a1, bx2, xC);
            }

            const int col = jt * 16 + l15;
            const float ba0 = bh[col],         bb0 = bx[col];
            const float ba1 = bh[D + col],     bb1 = bx[D + col];
            const float ba2 = bh[2 * D + col], bb2 = bx[2 * D + col];
#pragma unroll
            for (int r = 0; r < 8; ++r) {
                const int row = mt * 16 + r + 8 * hi;
                const float u  = sigmoidf_dev((hA[r] + ba0) + (xA[r] + bb0));
                const float rg = sigmoidf_dev((hB[r] + ba1) + (xB[r] + bb1));
                const float cd = tanhf_dev((xC[r] + bb2) + rg * (hC[r] + ba2));
                const float ho = h2[row * D + col];
                nv2[gi][r] = u * ho + (1.0f - u) * cd;
            }
        }
        __syncthreads();
#pragma unroll
        for (int gi = 0; gi < 2; ++gi) {
            const int g = wave + gi * 32;
            const int mt = g >> 3, jt = g & 7;
            const int col = jt * 16 + l15;
#pragma unroll
            for (int r = 0; r < 8; ++r)
                h2[(mt * 16 + r + 8 * hi) * D + col] = nv2[gi][r];
        }
        __syncthreads();
    }

    // outputs: combined (=h2 final) then stack([h1,h2], axis=-1) interleaved
    for (int i = tid; i < D * D; i += 1024) {
        const float a = h1[i];
        const float b = h2[i];
        out[i] = b;
        out[D * D + 2 * i]     = a;
        out[D * D + 2 * i + 1] = b;
    }
}

extern "C" void kernel_launch(void* const* d_in, const int* in_sizes, int n_in,
                              void* d_out, int out_size, void* d_ws, size_t ws_size,
                              hipStream_t stream) {
    const float* xs   = (const float*)d_in[0];   // (B,S,D) -> (1024,128)
    const int*   mask = (const int*)  d_in[1];   // (B,S)   -> (1024,)
    const float* Wh   = (const float*)d_in[2];   // (384,128)
    const float* bh   = (const float*)d_in[3];   // (384,)
    const float* Wx   = (const float*)d_in[4];   // (384,128)
    const float* bx   = (const float*)d_in[5];   // (384,)
    float* Xp  = (float*)d_ws;                   // (1024,384) = 1.5 MB scratch
    float* out = (float*)d_out;                  // 16384 + 32768 floats

    // Phase 1: token projections for layer 1, fully parallel (1536 WMMA tiles)
    gru_xproj_kernel<<<192, 256, 0, stream>>>(xs, Wx, bx, Xp);
    // Phase 2: fused 2-layer serial recurrence, one persistent workgroup
    gru_serial_kernel<<<1, 1024, 0, stream>>>(mask, Wh, bh, Wx, bx, Xp, out);
}